// HodgkinHuxley_8804682957011
// MI455X (gfx1250) — compile-verified
//
#include <hip/hip_runtime.h>
#include <cstdint>
#include <cstddef>

#define AS1 __attribute__((address_space(1)))
#define AS3 __attribute__((address_space(3)))

// ---- CDNA5 feature detection (device pass has amdgcn builtins; host pass falls back) ----
#if __has_builtin(__builtin_amdgcn_global_load_async_to_lds_b32)
#define HAVE_ASYNC_LDS 1
#else
#define HAVE_ASYNC_LDS 0
#endif

#if __has_builtin(__builtin_amdgcn_s_wait_asynccnt)
#define WAIT_ASYNC(n) __builtin_amdgcn_s_wait_asynccnt(n)
#else
#define WAIT_ASYNC(n) asm volatile("s_wait_asynccnt %0" ::"n"(n) : "memory")
#endif

#if __has_builtin(__builtin_amdgcn_exp2f)
#define EXP2(x) __builtin_amdgcn_exp2f(x)
#else
#define EXP2(x) exp2f(x)
#endif

#if __has_builtin(__builtin_amdgcn_rcpf)
#define RCP(x) __builtin_amdgcn_rcpf(x)
#else
#define RCP(x) (1.0f / (x))
#endif

#if __has_builtin(__builtin_nontemporal_store)
#define NTS(v, p) __builtin_nontemporal_store((v), (p))
#else
#define NTS(v, p) (*(p) = (v))
#endif

__global__ __launch_bounds__(256)
void hh_scan_kernel(const float* __restrict__ i_inj,   // [T, N]
                    const float* __restrict__ x0,      // [7, N]
                    const float* __restrict__ params,  // [28]
                    float* __restrict__ out,           // [T, 7, N]
                    int N, int T)
{
    constexpr int BLOCK = 256;
    constexpr int DEPTH = 8;  // async prefetch depth (LDS ring)

#if HAVE_ASYNC_LDS
    __shared__ float sbuf[DEPTH][BLOCK];
#endif

    const int tid = threadIdx.x;
    const int n   = blockIdx.x * BLOCK + tid;
    if (n >= N) return;

    // ---- constants (uniform loads, hoisted out of the T loop) ----
    const float dt = 0.1f;
    const float LOG2E = 1.44269504088896340736f;

    const float decay_ca = params[0],  rho_ca = params[1];
    const float p_tau = params[2],  p_scale = params[3],  p_mdp = params[4];
    const float q_tau = params[5],  q_scale = params[6],  q_mdp = params[7];
    const float n_tau = params[8],  n_scale = params[9],  n_mdp = params[10];
    const float f_tau = params[11], f_scale = params[12], f_mdp = params[13];
    const float e_tau = params[14], e_scale = params[15], e_mdp = params[16];
    const float h_alpha = params[17], h_scale = params[18], h_mdp = params[19];
    const float C_m = params[20], g_Ca = params[21], g_Ks = params[22];
    const float g_Kf = params[23], g_L = params[24];
    const float E_Ca = params[25], E_K = params[26], E_L = params[27];

    // gate(x,V,r) = A*(x/dt) + A*(inf/tau), A = tau*dt/(tau+dt)
    const float Ap = p_tau * dt / (p_tau + dt), Bp = Ap / dt, Cp = Ap / p_tau, cp = LOG2E / p_scale;
    const float Aq = q_tau * dt / (q_tau + dt), Bq = Aq / dt, Cq = Aq / q_tau, cq = LOG2E / q_scale;
    const float An = n_tau * dt / (n_tau + dt), Bn = An / dt, Cn = An / n_tau, cn = LOG2E / n_scale;
    const float Af = f_tau * dt / (f_tau + dt), Bf = Af / dt, Cf = Af / f_tau, cf = LOG2E / f_scale;
    const float Ae = e_tau * dt / (e_tau + dt), Be = Ae / dt, Ce = Ae / e_tau, ce = LOG2E / e_scale;
    const float ch = LOG2E / h_scale;
    const float inv_decay = 1.0f / decay_ca;
    const float dtC = dt / C_m;

    // ---- state in registers for the entire scan ----
    float V   = x0[0 * (size_t)N + n];
    float p   = x0[1 * (size_t)N + n];
    float q   = x0[2 * (size_t)N + n];
    float nn  = x0[3 * (size_t)N + n];
    float e   = x0[4 * (size_t)N + n];
    float f   = x0[5 * (size_t)N + n];
    float cac = x0[6 * (size_t)N + n];

    const float* gin = i_inj + n;

    auto step_store = [&](int t, float cur) {
        // h uses old cac
        float hs = RCP(1.0f + EXP2((h_mdp - cac) * ch));
        float h  = 1.0f + (hs - 1.0f) * h_alpha;
        float gce  = g_Ca * e * e * f * h;
        float vK   = V - E_K;
        float i_ca = gce * (V - E_Ca);
        float i_ks = g_Ks * nn * vK;
        float p2   = p * p;
        float i_kf = g_Kf * p2 * p2 * q * vK;
        float i_l  = g_L * (V - E_L);
        V = V + (cur - i_ca - i_ks - i_kf - i_l) * dtC;
        // cac uses I_Ca recomputed with the *updated* V
        float i_ca2 = gce * (V - E_Ca);
        cac = cac + (-i_ca2 * rho_ca - cac * inv_decay) * dt;
        // gates with updated V
        p  = Bp * p  + Cp * RCP(1.0f + EXP2((p_mdp - V) * cp));
        q  = Bq * q  + Cq * RCP(1.0f + EXP2((q_mdp - V) * cq));
        e  = Be * e  + Ce * RCP(1.0f + EXP2((e_mdp - V) * ce));
        f  = Bf * f  + Cf * RCP(1.0f + EXP2((f_mdp - V) * cf));
        nn = Bn * nn + Cn * RCP(1.0f + EXP2((n_mdp - V) * cn));
        // streaming NT stores: out[t][s][n], s = V,p,q,n,e,f,cac
        float* o = out + ((size_t)t * 7) * (size_t)N + n;
        NTS(V,   o);
        NTS(p,   o + 1 * (size_t)N);
        NTS(q,   o + 2 * (size_t)N);
        NTS(nn,  o + 3 * (size_t)N);
        NTS(e,   o + 4 * (size_t)N);
        NTS(f,   o + 5 * (size_t)N);
        NTS(cac, o + 6 * (size_t)N);
    };

#if HAVE_ASYNC_LDS
    // prologue: fill the async ring DEPTH steps deep
#pragma unroll
    for (int k = 0; k < DEPTH; ++k) {
        if (k < T) {
            __builtin_amdgcn_global_load_async_to_lds_b32(
                (AS1 int*)(gin + (size_t)k * N),
                (AS3 int*)(&sbuf[k][tid]), 0, 0);
        }
    }

    int t = 0;
    const int Tmain = (T > DEPTH) ? (T - DEPTH) : 0;
    for (; t < Tmain; ++t) {
        WAIT_ASYNC(DEPTH - 1);          // loads complete in order -> load t is done
        const int slot = t & (DEPTH - 1);
        float cur = sbuf[slot][tid];
        asm volatile("s_wait_dscnt 0" ::: "memory");  // drain LDS read before slot reuse
        __builtin_amdgcn_global_load_async_to_lds_b32(
            (AS1 int*)(gin + (size_t)(t + DEPTH) * N),
            (AS3 int*)(&sbuf[slot][tid]), 0, 0);
        step_store(t, cur);
    }
    for (; t < T; ++t) {                // tail: fewer than DEPTH outstanding
        WAIT_ASYNC(0);
        float cur = sbuf[t & (DEPTH - 1)][tid];
        step_store(t, cur);
    }
#else
    // fallback: direct loads + deep hardware prefetch (global_prefetch_b8)
    for (int t = 0; t < T; ++t) {
        if (t + 16 < T) __builtin_prefetch(gin + (size_t)(t + 16) * N, 0, 0);
        float cur = gin[(size_t)t * N];
        step_store(t, cur);
    }
#endif
}

extern "C" void kernel_launch(void* const* d_in, const int* in_sizes, int n_in,
                              void* d_out, int out_size, void* d_ws, size_t ws_size,
                              hipStream_t stream) {
    (void)n_in; (void)out_size; (void)d_ws; (void)ws_size;
    const float* i_inj  = (const float*)d_in[0];   // [T, N]
    const float* x0     = (const float*)d_in[1];   // [7, N]
    const float* params = (const float*)d_in[2];   // [28]
    float* out = (float*)d_out;                    // [T, 7, N]

    const int N = in_sizes[1] / 7;
    const int T = in_sizes[0] / N;
    const int blocks = (N + 255) / 256;

    hh_scan_kernel<<<blocks, 256, 0, stream>>>(i_inj, x0, params, out, N, T);
}